// TinyTCNInt8_51711406244184
// MI455X (gfx1250) — compile-verified
//
#include <hip/hip_runtime.h>

// ---------------------------------------------------------------------------
// TinyTCN on MI455X (gfx1250, wave32).
// Only the last time step feeds the head; receptive field = 121 samples, so we
// stage a 148-sample window per batch element in LDS (b128 transfers) and run
// the four dilated blocks as 16x16 WMMA tiles with fp32 V_WMMA_F32_16X16X4_F32
// (K in chunks of 4), full fp32 semantics. One wave = one batch element.
// Per-block tile ranges are pruned to the exact validity chain ending at the
// final sample; all im2col LDS offsets are hoisted to per-lane register tables
// (select of two compile-time constants on lane-half), so the inner tile loops
// are pure ds_load / v_wmma / ds_store with immediate tile offsets.
// ---------------------------------------------------------------------------

typedef float v2f __attribute__((ext_vector_type(2)));
typedef float v8f __attribute__((ext_vector_type(8)));

namespace {
constexpr int BW     = 256;        // LDS buffer width (time)
constexpr int PADJ   = 64;         // tile origin
constexpr int NTILES = 12;         // tiles t=0..11, tb = 64+16t
constexpr int WIN    = 148;        // staged real history (block0 reads >= j108)
constexpr int JX0    = BW - WIN;   // 108: x placed at j in [108,256), 16B-aligned
constexpr int TLEN   = 4096;
constexpr int NCH0   = 9;
}

#if __has_builtin(__builtin_amdgcn_wmma_f32_16x16x4_f32)
#define TCN_USE_WMMA 1
#else
#define TCN_USE_WMMA 0
#endif

// One TCN block:  y = relu( relu(conv2(relu(conv1(x)+b1))+b2) + wd*x + bd )
// bufX: block input; output written in place (residual is pointwise-safe).
// bufH: h1 scratch. P1T0/P2T0: first tile index of phase1 / phase2.
template <int CIN, int COUT, int DIL, int P1T0, int P2T0>
__device__ __forceinline__ void tcn_block(
    float* bufX, float* bufH,
    const float* __restrict__ w1, const float* __restrict__ b1,
    const float* __restrict__ w2, const float* __restrict__ b2,
    const float* __restrict__ wd, const float* __restrict__ bd)
{
  const int lane = (int)threadIdx.x;
#if TCN_USE_WMMA
  const int lo   = lane & 15;   // M (A-row) / N (B-col = time) index
  const int half = lane >> 4;   // selects K pair (A,B) / +8 C-row offset
  constexpr int K1 = CIN * 5;  constexpr int NK1 = (K1 + 3) / 4;
  constexpr int K2 = COUT * 5; constexpr int NK2 = (K2 + 3) / 4;
  constexpr int KD = CIN;      constexpr int NKD = (KD + 3) / 4;

  // conv im2col offset: kk = c*5+k -> (kk/5)*BW - (4-kk%5)*DIL, + lo.
  // After unrolling, j/e are literals; only the select on `half` is runtime.
  auto boffc = [&](int jj, int ee) -> int {
    const int kl = 4 * jj + ee, kh = 4 * jj + 2 + ee;
    const int o0 = (kl / 5) * BW - (4 - (kl % 5)) * DIL;
    const int o1 = (kh / 5) * BW - (4 - (kh % 5)) * DIL;
    return (half ? o1 : o0) + lo;
  };
  auto boffd = [&](int jj, int ee) -> int {  // pointwise: row kk, same column
    const int kl = 4 * jj + ee, kh = 4 * jj + 2 + ee;
    return (half ? kh : kl) * BW + lo;
  };
  const int srow = 8 * half * BW + lo;  // C/D store base (row r adds r*BW)

  // ---------------- phase 1 : h1 = relu(conv1(x) + b1) -> bufH -------------
  {
    v2f A[NK1];
    int ofb[NK1][2];
#pragma unroll
    for (int j = 0; j < NK1; ++j) {
      const int kk = 4 * j + 2 * half;
      A[j].x = (lo < COUT && kk     < K1) ? w1[lo * K1 + kk    ] : 0.0f;
      A[j].y = (lo < COUT && kk + 1 < K1) ? w1[lo * K1 + kk + 1] : 0.0f;
      ofb[j][0] = boffc(j, 0);
      ofb[j][1] = boffc(j, 1);
    }
    float bias[8];
#pragma unroll
    for (int r = 0; r < 8; ++r) {
      const int M = r + 8 * half;
      bias[r] = (M < COUT) ? b1[M] : 0.0f;
    }
#pragma unroll
    for (int t = P1T0; t < NTILES; ++t) {
      const int tb = PADJ + 16 * t;
      v8f acc = {};
#pragma unroll
      for (int j = 0; j < NK1; ++j) {
        v2f bv;
        bv.x = bufX[ofb[j][0] + tb];
        bv.y = bufX[ofb[j][1] + tb];
        acc = __builtin_amdgcn_wmma_f32_16x16x4_f32(false, A[j], false, bv,
                                                    (short)0, acc, false, false);
      }
#pragma unroll
      for (int r = 0; r < 8; ++r) {
        const float v = acc[r] + bias[r];
        bufH[srow + r * BW + tb] = v > 0.0f ? v : 0.0f;  // all 16 rows stored
      }
    }
  }
  __syncthreads();

  // ------ phase 2 : y = relu(relu(conv2(h1)+b2) + wd*x + bd) -> bufX -------
  {
    v2f A2[NK2], AD[NKD];
    int ofb2[NK2][2], ofbd[NKD][2];
#pragma unroll
    for (int j = 0; j < NK2; ++j) {
      const int kk = 4 * j + 2 * half;
      A2[j].x = (lo < COUT && kk     < K2) ? w2[lo * K2 + kk    ] : 0.0f;
      A2[j].y = (lo < COUT && kk + 1 < K2) ? w2[lo * K2 + kk + 1] : 0.0f;
      ofb2[j][0] = boffc(j, 0);
      ofb2[j][1] = boffc(j, 1);
    }
#pragma unroll
    for (int j = 0; j < NKD; ++j) {
      const int kk = 4 * j + 2 * half;
      AD[j].x = (lo < COUT && kk     < KD) ? wd[lo * KD + kk    ] : 0.0f;
      AD[j].y = (lo < COUT && kk + 1 < KD) ? wd[lo * KD + kk + 1] : 0.0f;
      ofbd[j][0] = boffd(j, 0);
      ofbd[j][1] = boffd(j, 1);
    }
    float bias2[8], biasd[8];
#pragma unroll
    for (int r = 0; r < 8; ++r) {
      const int M = r + 8 * half;
      bias2[r] = (M < COUT) ? b2[M] : 0.0f;
      biasd[r] = (M < COUT) ? bd[M] : 0.0f;
    }
#pragma unroll
    for (int t = P2T0; t < NTILES; ++t) {
      const int tb = PADJ + 16 * t;
      v8f c2 = {};
#pragma unroll
      for (int j = 0; j < NK2; ++j) {
        v2f bv;
        bv.x = bufH[ofb2[j][0] + tb];
        bv.y = bufH[ofb2[j][1] + tb];
        c2 = __builtin_amdgcn_wmma_f32_16x16x4_f32(false, A2[j], false, bv,
                                                   (short)0, c2, false, false);
      }
      v8f cr = {};  // 1x1 residual path, pointwise B from bufX
#pragma unroll
      for (int j = 0; j < NKD; ++j) {
        v2f bv;
        bv.x = bufX[ofbd[j][0] + tb];
        bv.y = bufX[ofbd[j][1] + tb];
        cr = __builtin_amdgcn_wmma_f32_16x16x4_f32(false, AD[j], false, bv,
                                                   (short)0, cr, false, false);
      }
#pragma unroll
      for (int r = 0; r < 8; ++r) {
        float o2 = c2[r] + bias2[r];
        o2 = o2 > 0.0f ? o2 : 0.0f;
        float y = o2 + cr[r] + biasd[r];
        y = y > 0.0f ? y : 0.0f;
        bufX[srow + r * BW + tb] = y;   // in place; reads done before stores
      }
    }
  }
  __syncthreads();
#else
  // -------- scalar fallback (host pass / missing builtin): same math -------
  for (int p = lane; p < BW - JX0; p += 32) {
    const int jj = JX0 + p;
    for (int o = 0; o < COUT; ++o) {
      float a = b1[o];
      for (int c = 0; c < CIN; ++c)
#pragma unroll
        for (int k = 0; k < 5; ++k)
          a += w1[(o * CIN + c) * 5 + k] * bufX[c * BW + jj - (4 - k) * DIL];
      bufH[o * BW + jj] = a > 0.0f ? a : 0.0f;
    }
  }
  __syncthreads();
  for (int p = lane; p < BW - JX0; p += 32) {
    const int jj = JX0 + p;
    float yv[16];
    for (int o = 0; o < COUT; ++o) {
      float a = b2[o];
      for (int c = 0; c < COUT; ++c)
#pragma unroll
        for (int k = 0; k < 5; ++k)
          a += w2[(o * COUT + c) * 5 + k] * bufH[c * BW + jj - (4 - k) * DIL];
      a = a > 0.0f ? a : 0.0f;
      float r = bd[o];
      for (int c = 0; c < CIN; ++c) r += wd[o * CIN + c] * bufX[c * BW + jj];
      const float y = a + r;
      yv[o] = y > 0.0f ? y : 0.0f;
    }
    for (int o = 0; o < COUT; ++o) bufX[o * BW + jj] = yv[o];
  }
  __syncthreads();
#endif
}

__global__ __launch_bounds__(32) void tcn_tiny_kernel(
    const float* __restrict__ x,
    const float* w1_0, const float* b1_0, const float* w2_0, const float* b2_0,
    const float* wd_0, const float* bd_0,
    const float* w1_1, const float* b1_1, const float* w2_1, const float* b2_1,
    const float* wd_1, const float* bd_1,
    const float* w1_2, const float* b1_2, const float* w2_2, const float* b2_2,
    const float* wd_2, const float* bd_2,
    const float* w1_3, const float* b1_3, const float* w2_3, const float* b2_3,
    const float* wd_3, const float* bd_3,
    const float* __restrict__ head_w, const float* __restrict__ head_b,
    float* __restrict__ out)
{
  __shared__ float bufX[16 * BW];
  __shared__ float bufH[16 * BW];
  const int b    = (int)blockIdx.x;
  const int lane = (int)threadIdx.x;

#if TCN_USE_WMMA
  // Zero only what the WMMA path reads as A=0 K-padding (NaN-propagation
  // safety): bufX rows 9..15, full width. bufH needs no init (every phase-2
  // read lands inside tiles phase 1 stored, all 16 rows), and bufX rows 0..8
  // below j=108 are never read (block0's deepest read is exactly j=108).
  {
    float4* z4 = (float4*)(bufX + 9 * BW);
    const float4 z = make_float4(0.0f, 0.0f, 0.0f, 0.0f);
    for (int i = lane; i < (7 * BW) / 4; i += 32) z4[i] = z;
  }
#else
  for (int i = lane; i < 16 * BW; i += 32) { bufX[i] = 0.0f; bufH[i] = 0.0f; }
#endif
  __syncthreads();

  // stage the last WIN=148 samples of the 9 input channels, b128 transfers
  // (global byte offset (4096-148)*4 and LDS offset (c*256+108)*4 are both
  //  16B-aligned; WIN/4 = 37 vectors per row)
  {
    constexpr int W4 = WIN / 4;
    const float4* xb4 =
        (const float4*)(x + (size_t)b * (NCH0 * TLEN) + (TLEN - WIN));
    for (int i = lane; i < NCH0 * W4; i += 32) {
      const int c = i / W4, j = i - c * W4;
      *(float4*)(bufX + c * BW + JX0 + 4 * j) = xb4[c * (TLEN / 4) + j];
    }
  }
  __syncthreads();

  // Tile-range pruning (tb = 64+16t), exact validity chain to final sample:
  //   b0: h1 tiles 3.. (j>=112 reads x>=108), y0 tiles 4.. (valid j>=128)
  //   b1: h1 tiles 4.. (valid j>=136),        y1 tiles 5.. (valid j>=144)
  //   b2: h1 tiles 6.. (valid j>=160),        y2 tiles 7.. (valid j>=176)
  //   b3: h1 tiles 9.. (valid j>=208),        y3 tile 11   (valid j>=240)
  tcn_block< 9, 11, 1, 3,  4>(bufX, bufH, w1_0, b1_0, w2_0, b2_0, wd_0, bd_0);
  tcn_block<11, 10, 2, 4,  5>(bufX, bufH, w1_1, b1_1, w2_1, b2_1, wd_1, bd_1);
  tcn_block<10,  5, 4, 6,  7>(bufX, bufH, w1_2, b1_2, w2_2, b2_2, wd_2, bd_2);
  tcn_block< 5,  4, 8, 9, 11>(bufX, bufH, w1_3, b1_3, w2_3, b2_3, wd_3, bd_3);

  if (lane == 0) {  // head on the final time step only
    float a = head_b[0];
#pragma unroll
    for (int c = 0; c < 4; ++c) a += head_w[c] * bufX[c * BW + (BW - 1)];
    out[b] = a;
  }
}

extern "C" void kernel_launch(void* const* d_in, const int* in_sizes, int n_in,
                              void* d_out, int out_size, void* d_ws, size_t ws_size,
                              hipStream_t stream) {
  (void)n_in; (void)out_size; (void)d_ws; (void)ws_size;
  const float* x = (const float*)d_in[0];
  const float* p[24];
  for (int i = 0; i < 24; ++i) p[i] = (const float*)d_in[1 + i];
  const float* head_w = (const float*)d_in[25];
  const float* head_b = (const float*)d_in[26];
  const int nb = in_sizes[0] / (NCH0 * TLEN);   // 2048 batch elements
  tcn_tiny_kernel<<<nb, 32, 0, stream>>>(
      x,
      p[0],  p[1],  p[2],  p[3],  p[4],  p[5],
      p[6],  p[7],  p[8],  p[9],  p[10], p[11],
      p[12], p[13], p[14], p[15], p[16], p[17],
      p[18], p[19], p[20], p[21], p[22], p[23],
      head_w, head_b, (float*)d_out);
}